// GNNModel_27900107555264
// MI455X (gfx1250) — compile-verified
//
#include <hip/hip_runtime.h>

typedef float v2f __attribute__((ext_vector_type(2)));
typedef float v8f __attribute__((ext_vector_type(8)));

#define F_IN 128
#define HDIM 64
#define CDIM 2
#define BN_EPS 1e-5f

// LDS tile strides padded to spread banks (64 banks x 4B on CDNA5):
// A row stride 132 -> lane r hits bank (4r+c)%64 (conflict-free b64 loads)
// B row stride 68  -> half-waves (k vs k+2) land 8 banks apart
#define A_STRIDE 132
#define B_STRIDE 68

// ---------------- zero fill ----------------
__global__ void k_zero(float* __restrict__ p, size_t n) {
    size_t i = (size_t)blockIdx.x * blockDim.x + threadIdx.x;
    if (i < n) p[i] = 0.0f;
}

// ---------------- degree (counted on dst) ----------------
__global__ void k_degree(const long long* __restrict__ dst, float* __restrict__ deg, int E) {
    int t = blockIdx.x * blockDim.x + threadIdx.x;
    if (t < E) atomicAdd(&deg[(int)dst[t]], 1.0f);
}

// deg -> dinv = rsqrt(deg + 1)   (+1 = self loop)
__global__ void k_dinv(float* __restrict__ deg, int N) {
    int i = blockIdx.x * blockDim.x + threadIdx.x;
    if (i < N) deg[i] = rsqrtf(deg[i] + 1.0f);
}

// ---------------- GEMM1: h = x @ W1  (fp32 WMMA 16x16x4) ----------------
// 4 waves/block = 4 column tiles of H=64 over one 16-row tile of x.
// x tile (16x128) and all of W1 (128x64) are staged into LDS with the CDNA5
// async global->LDS DMA path; the K-loop then runs entirely out of LDS.
__global__ __launch_bounds__(128) void k_gemm1_wmma(
        const float* __restrict__ x, const float* __restrict__ W1,
        float* __restrict__ h, int N) {
    __shared__ float sA[16 * A_STRIDE];       // padded 16x128 x-tile
    __shared__ float sB[F_IN * B_STRIDE];     // padded 128x64 W1

    const int tid  = threadIdx.x;
    const int lane = tid & 31;
    const int wave = tid >> 5;                // 0..3 -> column tile
    const int row0 = blockIdx.x * 16;
    if (row0 >= N) return;                    // block-uniform

    // ---- async stage x tile: 512 x 16B chunks (4 per thread)
    {
        const unsigned ldsA = (unsigned)(size_t)(&sA[0]);
        #pragma unroll
        for (int i = 0; i < 4; ++i) {
            int c    = tid + 128 * i;         // 0..511
            int rrow = c >> 5;                // 32 chunks per 128-float row
            int rcol = (c & 31) * 4;
            int grow = row0 + rrow;
            if (grow >= N) grow = N - 1;      // clamp: EXEC stays full
            const float* gp = x + (size_t)grow * F_IN + rcol;
            unsigned lp = ldsA + (unsigned)(rrow * A_STRIDE + rcol) * 4u;
            asm volatile("global_load_async_to_lds_b128 %0, %1, off"
                         :: "v"(lp), "v"(gp) : "memory");
        }
    }
    // ---- async stage W1: 2048 x 16B chunks (16 per thread)
    {
        const unsigned ldsB = (unsigned)(size_t)(&sB[0]);
        #pragma unroll
        for (int i = 0; i < 16; ++i) {
            int c    = tid + 128 * i;         // 0..2047
            int rrow = c >> 4;                // 16 chunks per 64-float row
            int rcol = (c & 15) * 4;
            const float* gp = W1 + (size_t)rrow * HDIM + rcol;
            unsigned lp = ldsB + (unsigned)(rrow * B_STRIDE + rcol) * 4u;
            asm volatile("global_load_async_to_lds_b128 %0, %1, off"
                         :: "v"(lp), "v"(gp) : "memory");
        }
    }
    asm volatile("s_wait_asynccnt 0x0" ::: "memory");
    __syncthreads();

    const int r     = lane & 15;              // A row / B,D column
    const int khalf = (lane >> 4) * 2;        // lanes 0-15: K=0,1  16-31: K=2,3
    const int col0  = wave * 16;

    v8f acc = {};
    #pragma unroll 4
    for (int k0 = 0; k0 < F_IN; k0 += 4) {
        v2f a, b;
        a.x = sA[r * A_STRIDE + k0 + khalf];
        a.y = sA[r * A_STRIDE + k0 + khalf + 1];
        b.x = sB[(k0 + khalf)     * B_STRIDE + col0 + r];
        b.y = sB[(k0 + khalf + 1) * B_STRIDE + col0 + r];
        acc = __builtin_amdgcn_wmma_f32_16x16x4_f32(
                false, a, false, b, (short)0, acc, false, false);
    }

    // D layout: element i -> (M = i + (lane>=16 ? 8 : 0), N = lane&15)
    const int mofs = (lane >> 4) ? 8 : 0;
    float* hp = h + (size_t)(row0 + mofs) * HDIM + col0 + r;
    if (row0 + 16 <= N) {                     // block-uniform fast path
        #pragma unroll
        for (int i = 0; i < 8; ++i) hp[(size_t)i * HDIM] = acc[i];
    } else {
        #pragma unroll
        for (int i = 0; i < 8; ++i)
            if (row0 + mofs + i < N) hp[(size_t)i * HDIM] = acc[i];
    }
}

// ---------------- message passing layer 1 ----------------
// 16 threads per edge, 4 feats each. Edge metadata is loaded by one lane per
// 16-lane group and broadcast with wave32 shuffles.
__global__ void k_mp1(const long long* __restrict__ src, const long long* __restrict__ dst,
                      const float* __restrict__ h, const float* __restrict__ dinv,
                      float* __restrict__ agg, int E) {
    int t = blockIdx.x * blockDim.x + threadIdx.x;
    int e = t >> 4;
    if (e >= E) return;
    int lane = threadIdx.x & 31;
    int base = lane & 16;                    // leader lane of my 16-lane group
    int s = 0, d = 0;
    float coef = 0.0f;
    if ((lane & 15) == 0) {
        s = (int)src[e];
        d = (int)dst[e];
        coef = dinv[s] * dinv[d];
    }
    s    = __shfl(s, base, 32);
    d    = __shfl(d, base, 32);
    coef = __shfl(coef, base, 32);

    int fg = (t & 15) * 4;
    const float4 hv = *(const float4*)(h + (size_t)s * HDIM + fg);
    float* bp = agg + (size_t)d * HDIM + fg;
    atomicAdd(bp + 0, hv.x * coef);
    atomicAdd(bp + 1, hv.y * coef);
    atomicAdd(bp + 2, hv.z * coef);
    atomicAdd(bp + 3, hv.w * coef);
}

// ---------------- finalize layer1 + batch stats ----------------
// out1 = agg + h*dinv^2 + b1 (in place); per-feature sum/sumsq -> stats via
// LDS reduce + one atomic per feature per block.
__global__ __launch_bounds__(256) void k_fin1_stats(
        float* __restrict__ agg, const float* __restrict__ h,
        const float* __restrict__ dinv, const float* __restrict__ b1,
        float* __restrict__ stats, int N) {
    __shared__ float s_sum[256];
    __shared__ float s_sq[256];
    const int f = threadIdx.x & (HDIM - 1);   // strides are multiples of 64
    float lsum = 0.0f, lsq = 0.0f;
    const size_t total = (size_t)N * HDIM;
    for (size_t idx = (size_t)blockIdx.x * blockDim.x + threadIdx.x; idx < total;
         idx += (size_t)gridDim.x * blockDim.x) {
        int node = (int)(idx >> 6);
        float dv = dinv[node];
        float v  = agg[idx] + h[idx] * dv * dv + b1[f];
        agg[idx] = v;
        lsum += v;
        lsq  += v * v;
    }
    s_sum[threadIdx.x] = lsum;
    s_sq[threadIdx.x]  = lsq;
    __syncthreads();
    if (threadIdx.x < HDIM) {
        float a = s_sum[threadIdx.x] + s_sum[threadIdx.x + 64] +
                  s_sum[threadIdx.x + 128] + s_sum[threadIdx.x + 192];
        float q = s_sq[threadIdx.x] + s_sq[threadIdx.x + 64] +
                  s_sq[threadIdx.x + 128] + s_sq[threadIdx.x + 192];
        atomicAdd(&stats[threadIdx.x], a);
        atomicAdd(&stats[HDIM + threadIdx.x], q);
    }
}

// ---------------- BN + ReLU + GEMM2 fused (h2 never materialized) ----------------
// Per-feature scale/shift + W2 prepared once per block in LDS.
__global__ __launch_bounds__(256) void k_bn_relu_gemm2(
        const float* __restrict__ out1, const float* __restrict__ stats,
        const float* __restrict__ gamma, const float* __restrict__ beta,
        const float* __restrict__ W2, float* __restrict__ g, int N) {
    __shared__ float s_scale[HDIM], s_shift[HDIM], s_w0[HDIM], s_w1[HDIM];
    if (threadIdx.x < HDIM) {
        int f = threadIdx.x;
        float invN = 1.0f / (float)N;
        float mean = stats[f] * invN;
        float var  = stats[HDIM + f] * invN - mean * mean;
        float sc   = gamma[f] * rsqrtf(var + BN_EPS);
        s_scale[f] = sc;
        s_shift[f] = beta[f] - mean * sc;
        s_w0[f]    = W2[f * CDIM + 0];
        s_w1[f]    = W2[f * CDIM + 1];
    }
    __syncthreads();
    int i = blockIdx.x * blockDim.x + threadIdx.x;
    if (i >= N) return;
    const float4* row = (const float4*)(out1 + (size_t)i * HDIM);
    float acc0 = 0.0f, acc1 = 0.0f;
    #pragma unroll
    for (int q = 0; q < HDIM / 4; ++q) {
        float4 v = row[q];
        int f = q * 4;
        float v0 = fmaxf(v.x * s_scale[f + 0] + s_shift[f + 0], 0.0f);
        float v1 = fmaxf(v.y * s_scale[f + 1] + s_shift[f + 1], 0.0f);
        float v2 = fmaxf(v.z * s_scale[f + 2] + s_shift[f + 2], 0.0f);
        float v3 = fmaxf(v.w * s_scale[f + 3] + s_shift[f + 3], 0.0f);
        acc0 += v0 * s_w0[f + 0] + v1 * s_w0[f + 1] + v2 * s_w0[f + 2] + v3 * s_w0[f + 3];
        acc1 += v0 * s_w1[f + 0] + v1 * s_w1[f + 1] + v2 * s_w1[f + 2] + v3 * s_w1[f + 3];
    }
    g[(size_t)i * CDIM + 0] = acc0;
    g[(size_t)i * CDIM + 1] = acc1;
}

// ---------------- message passing layer 2 ----------------
__global__ void k_mp2(const long long* __restrict__ src, const long long* __restrict__ dst,
                      const float* __restrict__ g, const float* __restrict__ dinv,
                      float* __restrict__ out, int E) {
    int e = blockIdx.x * blockDim.x + threadIdx.x;
    if (e >= E) return;
    int s = (int)src[e], d = (int)dst[e];
    float coef = dinv[s] * dinv[d];
    atomicAdd(out + (size_t)d * CDIM + 0, g[(size_t)s * CDIM + 0] * coef);
    atomicAdd(out + (size_t)d * CDIM + 1, g[(size_t)s * CDIM + 1] * coef);
}

// self-loop contribution + bias for layer 2
__global__ void k_self2(const float* __restrict__ g, const float* __restrict__ dinv,
                        const float* __restrict__ b2, float* __restrict__ out, int N) {
    int i = blockIdx.x * blockDim.x + threadIdx.x;
    if (i >= N) return;
    float dv = dinv[i];
    float c  = dv * dv;
    out[(size_t)i * CDIM + 0] += g[(size_t)i * CDIM + 0] * c + b2[0];
    out[(size_t)i * CDIM + 1] += g[(size_t)i * CDIM + 1] * c + b2[1];
}

extern "C" void kernel_launch(void* const* d_in, const int* in_sizes, int n_in,
                              void* d_out, int out_size, void* d_ws, size_t ws_size,
                              hipStream_t stream) {
    const float*     x     = (const float*)d_in[0];
    const long long* ei    = (const long long*)d_in[1];   // int64 [2, E]
    const float*     W1    = (const float*)d_in[2];
    const float*     b1    = (const float*)d_in[3];
    const float*     gamma = (const float*)d_in[4];
    const float*     beta  = (const float*)d_in[5];
    const float*     W2    = (const float*)d_in[6];
    const float*     b2    = (const float*)d_in[7];

    const int N = in_sizes[0] / F_IN;
    const int E = in_sizes[1] / 2;
    const long long* src = ei;
    const long long* dst = ei + E;

    // workspace layout (floats, 64-element aligned blocks):
    // dinv[NA] | h[NA*64] | agg[NA*64] | g[NA*2] | stats[128]
    const size_t NA = ((size_t)N + 63) & ~(size_t)63;
    float* ws    = (float*)d_ws;
    float* dinv  = ws;
    float* h     = dinv + NA;
    float* agg   = h + NA * HDIM;
    float* g     = agg + NA * HDIM;
    float* stats = g + NA * CDIM;
    float* out   = (float*)d_out;

    const int T = 256;
    // zero: degree accumulator, agg, stats, d_out (every call -> replay-safe)
    k_zero<<<(int)((NA + T - 1) / T), T, 0, stream>>>(dinv, NA);
    {
        size_t n = (size_t)N * HDIM;
        k_zero<<<(int)((n + T - 1) / T), T, 0, stream>>>(agg, n);
    }
    k_zero<<<1, 128, 0, stream>>>(stats, (size_t)128);
    {
        size_t n = (size_t)N * CDIM;
        k_zero<<<(int)((n + T - 1) / T), T, 0, stream>>>(out, n);
    }

    // degree + norm
    k_degree<<<(E + T - 1) / T, T, 0, stream>>>(dst, dinv, E);
    k_dinv<<<(N + T - 1) / T, T, 0, stream>>>(dinv, N);

    // GEMM1: h = x @ W1 (fp32 WMMA, async-LDS staged)
    k_gemm1_wmma<<<(N + 15) / 16, 128, 0, stream>>>(x, W1, h, N);

    // message passing layer 1 (16 threads/edge)
    {
        long long tt = (long long)E * 16;
        k_mp1<<<(int)((tt + T - 1) / T), T, 0, stream>>>(src, dst, h, dinv, agg, E);
    }

    // finalize out1 + batch stats
    k_fin1_stats<<<512, 256, 0, stream>>>(agg, h, dinv, b1, stats, N);

    // BN + ReLU + GEMM2 fused
    k_bn_relu_gemm2<<<(N + T - 1) / T, T, 0, stream>>>(agg, stats, gamma, beta, W2, g, N);

    // message passing layer 2 + self loop/bias
    k_mp2<<<(E + T - 1) / T, T, 0, stream>>>(src, dst, g, dinv, out, E);
    k_self2<<<(N + T - 1) / T, T, 0, stream>>>(g, dinv, b2, out, N);
}